// PRNNCell_858993459383
// MI455X (gfx1250) — compile-verified
//
#include <hip/hip_runtime.h>
#include <math.h>

// Problem constants
#define BB 32
#define II 1024
#define HH 1024
#define DD 2048          // I + H
#define NG 2048          // 2*H (gate width)
#define ALPHA 0.001f
#define BETA  0.01f

typedef __attribute__((ext_vector_type(2))) float v2f;
typedef __attribute__((ext_vector_type(4))) float v4f;
typedef __attribute__((ext_vector_type(8))) float v8f;

// ---------------------------------------------------------------------------
// Kernel A: gate = sigmoid(x @ Wg^T + bg), x = [inputs | hidden]  (32 x 2048)
// f32 WMMA 16x16x4. One wave computes a 16(M) x 64(N) tile (4 accumulators),
// K loop split into inputs-phase (K 0..1023) and hidden-phase (K 1024..2047).
// A-fragment layout (32-bit 16x4): lane L holds M = L%16, K = 2*(L>=16)+v.
// B-fragment (4x16) symmetric: lane L holds N = L%16, K = 2*(L>=16)+v.
// ---------------------------------------------------------------------------
__global__ __launch_bounds__(256) void gate_gemm_wmma(
    const float* __restrict__ inputs, const float* __restrict__ hidden,
    const float* __restrict__ Wg, const float* __restrict__ bg,
    float* __restrict__ gate)
{
    const int lane   = threadIdx.x & 31;
    const int wave   = threadIdx.x >> 5;
    const int waveId = blockIdx.x * 8 + wave;        // 0..63
    const int mtile  = (waveId & 1) << 4;            // 0 or 16
    const int ntile0 = (waveId >> 1) << 6;           // 0..1984 step 64
    const int half   = lane >> 4;
    const int lm     = lane & 15;

    v8f acc[4] = {};

    const int m = mtile + lm;                        // row of x this lane feeds
    const float* Arow[2] = { inputs + m * II, hidden + m * HH };

    for (int phase = 0; phase < 2; ++phase) {
        const float* __restrict__ ar = Arow[phase];
        const int kofs = phase * 1024;
        for (int k = 0; k < 1024; k += 4) {
            const int kk = k + 2 * half;
            v2f a;
            a.x = ar[kk];
            a.y = ar[kk + 1];
#pragma unroll
            for (int s = 0; s < 4; ++s) {
                const int n = ntile0 + 16 * s + lm;  // row of Wg (col of gate)
                const float* __restrict__ br = Wg + (size_t)n * DD + kofs;
                v2f bf;
                bf.x = br[kk];
                bf.y = br[kk + 1];
                acc[s] = __builtin_amdgcn_wmma_f32_16x16x4_f32(
                    false, a, false, bf, (short)0, acc[s], false, false);
            }
        }
    }

    // C/D layout: VGPR r, lanes 0-15 -> M=r, lanes 16-31 -> M=8+r; N = lane%16
#pragma unroll
    for (int s = 0; s < 4; ++s) {
        const int n = ntile0 + 16 * s + lm;
        const float bgn = bg[n];
#pragma unroll
        for (int r = 0; r < 8; ++r) {
            const int mr = mtile + r + 8 * half;     // 0..31
            float v = acc[s][r] + bgn;
            gate[mr * NG + n] = 1.0f / (1.0f + __expf(-v));
        }
    }
}

// ---------------------------------------------------------------------------
// Kernel B: fused matvec + tanh + fast-weight update, one block per (b,k) row.
// wih row (2048 f32 = 8KB) loaded ONCE into registers (2 x b128 / thread) with
// NT temporal hint (zero reuse — keep it out of L2), dot-reduced, then updated
// and streamed back out with NT stores. Wih_a..d (32MB, reused 32x across the
// batch) use default temporal policy so they stay L2-resident (192MB L2).
// ---------------------------------------------------------------------------
__global__ __launch_bounds__(256) void prnn_row_update(
    const float* __restrict__ inputs, const float* __restrict__ hidden,
    const float* __restrict__ wih,
    const float* __restrict__ Wa, const float* __restrict__ Wb,
    const float* __restrict__ Wc, const float* __restrict__ Wd,
    const float* __restrict__ bih, const float* __restrict__ gate,
    float* __restrict__ nh_out, float* __restrict__ wih_out)
{
    __shared__ __align__(16) float xs[DD];
    __shared__ float red[8];
    __shared__ float bc[3];

    const int bk   = blockIdx.x;        // b*H + k
    const int b    = bk >> 10;
    const int k    = bk & 1023;
    const int t    = threadIdx.x;
    const int lane = t & 31;
    const int wv   = t >> 5;

    // stage x[b] = [inputs[b] | hidden[b]] into LDS (8 KB)
    const float* __restrict__ inb = inputs + b * II;
    const float* __restrict__ hb  = hidden + b * HH;
    for (int i = t; i < II; i += 256) xs[i] = inb[i];
    for (int i = t; i < HH; i += 256) xs[II + i] = hb[i];
    __syncthreads();

    // load this wih row: 8 consecutive floats per thread (two NT b128 loads)
    const size_t rowoff = (size_t)bk * DD;
    const v4f* __restrict__ wrow4 = (const v4f*)(wih + rowoff);
    const int j4 = t * 2;                 // v4f index; covers j = t*8..t*8+7
    const v4f w0 = __builtin_nontemporal_load(wrow4 + j4);
    const v4f w1 = __builtin_nontemporal_load(wrow4 + j4 + 1);
    const v4f* xs4 = (const v4f*)xs;
    const v4f x0 = xs4[j4];
    const v4f x1 = xs4[j4 + 1];

    float partial = w0.x * x0.x + w0.y * x0.y + w0.z * x0.z + w0.w * x0.w
                  + w1.x * x1.x + w1.y * x1.y + w1.z * x1.z + w1.w * x1.w;
#pragma unroll
    for (int off = 16; off > 0; off >>= 1)
        partial += __shfl_down(partial, off, 32);
    if (lane == 0) red[wv] = partial;
    __syncthreads();

    if (t == 0) {
        float dot = 0.0f;
#pragma unroll
        for (int w = 0; w < 8; ++w) dot += red[w];
        const float y  = tanhf(dot + bih[k]);
        const float fg = gate[b * NG + k];          // forget gate
        const float ig = gate[b * NG + HH + k];     // input gate
        nh_out[bk] = y;
        bc[0] = y;
        bc[1] = 1.0f - BETA * fg;                   // decay
        bc[2] = ALPHA * ig;                         // learning-rate scale
    }
    __syncthreads();

    const float y     = bc[0];
    const float decay = bc[1];
    const float lr    = bc[2];

    const size_t koff = (size_t)k * DD;
    const v4f* __restrict__ A4 = (const v4f*)(Wa + koff);
    const v4f* __restrict__ B4 = (const v4f*)(Wb + koff);
    const v4f* __restrict__ C4 = (const v4f*)(Wc + koff);
    const v4f* __restrict__ D4 = (const v4f*)(Wd + koff);
    const v4f a0 = A4[j4], a1 = A4[j4 + 1];
    const v4f bb0 = B4[j4], bb1 = B4[j4 + 1];
    const v4f c0 = C4[j4], c1 = C4[j4 + 1];
    const v4f d0 = D4[j4], d1 = D4[j4 + 1];

    v4f o0, o1;
    // update = y*x*a + x*b + y*c + d ; out = w*decay + lr*update
    o0.x = w0.x * decay + lr * (y * x0.x * a0.x + x0.x * bb0.x + y * c0.x + d0.x);
    o0.y = w0.y * decay + lr * (y * x0.y * a0.y + x0.y * bb0.y + y * c0.y + d0.y);
    o0.z = w0.z * decay + lr * (y * x0.z * a0.z + x0.z * bb0.z + y * c0.z + d0.z);
    o0.w = w0.w * decay + lr * (y * x0.w * a0.w + x0.w * bb0.w + y * c0.w + d0.w);
    o1.x = w1.x * decay + lr * (y * x1.x * a1.x + x1.x * bb1.x + y * c1.x + d1.x);
    o1.y = w1.y * decay + lr * (y * x1.y * a1.y + x1.y * bb1.y + y * c1.y + d1.y);
    o1.z = w1.z * decay + lr * (y * x1.z * a1.z + x1.z * bb1.z + y * c1.z + d1.z);
    o1.w = w1.w * decay + lr * (y * x1.w * a1.w + x1.w * bb1.w + y * c1.w + d1.w);

    v4f* __restrict__ out4 = (v4f*)(wih_out + rowoff);
    __builtin_nontemporal_store(o0, out4 + j4);
    __builtin_nontemporal_store(o1, out4 + j4 + 1);
}

extern "C" void kernel_launch(void* const* d_in, const int* in_sizes, int n_in,
                              void* d_out, int out_size, void* d_ws, size_t ws_size,
                              hipStream_t stream) {
    const float* inputs = (const float*)d_in[0];   // (32,1024)
    const float* hidden = (const float*)d_in[1];   // (32,1024)
    const float* wih    = (const float*)d_in[2];   // (32,1024,2048)
    const float* Wa     = (const float*)d_in[3];   // (1024,2048)
    const float* Wb     = (const float*)d_in[4];
    const float* Wc     = (const float*)d_in[5];
    const float* Wd     = (const float*)d_in[6];
    const float* bih    = (const float*)d_in[7];   // (1024,)
    const float* Wg     = (const float*)d_in[8];   // (2048,2048)
    const float* bg     = (const float*)d_in[9];   // (2048,)

    float* out      = (float*)d_out;
    float* nh       = out;                 // new_hidden: 32*1024
    float* wih_new  = out + BB * HH;       // wih_new: 32*1024*2048
    float* gate     = (float*)d_ws;        // (32,2048) f32 = 256 KB scratch

    // 64 wave-tiles (2 M-tiles x 32 N-tiles of 64) / 8 waves per block = 8 blocks
    gate_gemm_wmma<<<8, 256, 0, stream>>>(inputs, hidden, Wg, bg, gate);

    // one block per (b,k) row of wih
    prnn_row_update<<<BB * HH, 256, 0, stream>>>(
        inputs, hidden, wih, Wa, Wb, Wc, Wd, bih, gate, nh, wih_new);
}